// SSGConv_936302871060
// MI455X (gfx1250) — compile-verified
//
#include <hip/hip_runtime.h>
#include <stdint.h>

#define CHANNELS 64
#define STEPS 10
#define ALPHA 0.1f
#define TILE 256     // CSR entries staged into LDS per block iteration
#define WPB 8        // waves (nodes) per block; block = 256 threads = 8 wave32

// ---- CDNA5 async global->LDS staging (CDNA5 ISA) ----
#if defined(__gfx1250__) && __has_builtin(__builtin_amdgcn_global_load_async_to_lds_b64)
#define USE_ASYNC_LDS 1
#else
#define USE_ASYNC_LDS 0
#endif

// b64 async copy moves one 8-byte element; builtin params are typed v2i pointers
typedef int v2i __attribute__((vector_size(8)));                 // 2 x i32
typedef __attribute__((address_space(1))) v2i gas_v2i;           // global (AS1)
typedef __attribute__((address_space(3))) v2i las_v2i;           // LDS (AS3)

__device__ __forceinline__ void wait_async0() {
#if defined(__gfx1250__)
#if __has_builtin(__builtin_amdgcn_s_wait_asynccnt)
  __builtin_amdgcn_s_wait_asynccnt(0);
#else
  asm volatile("s_wait_asynccnt 0" ::: "memory");
#endif
#endif
}

// ---------------- utility kernels ----------------
__global__ void zero_i32(int* __restrict__ p, int n) {
  int i = blockIdx.x * blockDim.x + threadIdx.x;
  if (i < n) p[i] = 0;
}

__global__ void zero_f32v4(float4* __restrict__ p, int n4) {
  int i = blockIdx.x * blockDim.x + threadIdx.x;
  if (i < n4) p[i] = make_float4(0.f, 0.f, 0.f, 0.f);
}

// degree count (int) + int64 -> int32 index conversion
__global__ void deg_convert_k(const long long* __restrict__ ei,
                              int* __restrict__ src32, int* __restrict__ dst32,
                              int* __restrict__ deg, int E) {
  int e = blockIdx.x * blockDim.x + threadIdx.x;
  if (e >= E) return;
  int s = (int)ei[e];
  int d = (int)ei[(size_t)E + e];
  src32[e] = s;
  dst32[e] = d;
  atomicAdd(&deg[d], 1);
}

__global__ void dinv_k(const int* __restrict__ deg, float* __restrict__ dinv, int n) {
  int i = blockIdx.x * blockDim.x + threadIdx.x;
  if (i < n) {
    float d = (float)deg[i];
    dinv[i] = 1.0f / sqrtf(fmaxf(d, 1.0f));
  }
}

// single-block exclusive scan of degrees -> off[0..n] and cursor copy
__global__ __launch_bounds__(1024) void scan_k(const int* __restrict__ deg,
                                               int* __restrict__ off,
                                               int* __restrict__ cursor, int n) {
  __shared__ int ssum[1024];
  const int t = threadIdx.x;
  const int per = (n + 1023) >> 10;
  int beg = t * per;
  int end = beg + per; if (end > n) end = n;
  int s = 0;
  for (int i = beg; i < end; ++i) s += deg[i];
  ssum[t] = s;
  __syncthreads();
  for (int d2 = 1; d2 < 1024; d2 <<= 1) {   // Hillis-Steele inclusive scan
    int v = (t >= d2) ? ssum[t - d2] : 0;
    __syncthreads();
    ssum[t] += v;
    __syncthreads();
  }
  int prefix = (t == 0) ? 0 : ssum[t - 1];
  for (int i = beg; i < end; ++i) {
    off[i] = prefix;
    cursor[i] = prefix;
    prefix += deg[i];
  }
  if (t == 0) off[n] = ssum[1023];
}

// scatter edges into CSR slots keyed by dst; entry = {src, bits(norm)}
__global__ void fill_k(const int* __restrict__ src32, const int* __restrict__ dst32,
                       const float* __restrict__ dinv, int* __restrict__ cursor,
                       int2* __restrict__ csr, int E) {
  int e = blockIdx.x * blockDim.x + threadIdx.x;
  if (e >= E) return;
  int s = src32[e], d = dst32[e];
  int slot = atomicAdd(&cursor[d], 1);
  float w = dinv[s] * dinv[d];
  csr[slot] = make_int2(s, __float_as_int(w));
}

// one propagation step: hnext[n] = sum_{e: dst==n} w_e * hcur[src_e]; acc += hnext
// atomic-free: wave per node, float2 per lane; block's contiguous CSR window is
// staged into LDS with async global->LDS copies (ASYNCcnt + s_wait_asynccnt).
__global__ __launch_bounds__(256) void step_k(const float* __restrict__ hcur,
                                              float* __restrict__ hnext,
                                              float* __restrict__ acc,
                                              const int* __restrict__ off,
                                              const int2* __restrict__ csr, int n) {
  __shared__ int2 s_e[TILE];
  const int tid  = threadIdx.x;
  const int wave = tid >> 5;
  const int lane = tid & 31;
  const int ch   = lane << 1;                 // 2 channels per lane
  const int nodeBase = blockIdx.x * WPB;
  if (nodeBase >= n) return;
  const int node  = nodeBase + wave;
  const bool valid = node < n;
  const int rowStart = valid ? off[node]     : 0;
  const int rowEnd   = valid ? off[node + 1] : 0;
  const int winLastN = (nodeBase + WPB < n) ? (nodeBase + WPB) : n;
  const int winStart = off[nodeBase];
  const int winEnd   = off[winLastN];

  float2 a; a.x = 0.0f; a.y = 0.0f;

  for (int tb = winStart; tb < winEnd; tb += TILE) {
    const int cnt = min(TILE, winEnd - tb);
    __syncthreads();                          // protect LDS reuse
    if (tid < cnt) {
#if USE_ASYNC_LDS
      __builtin_amdgcn_global_load_async_to_lds_b64(
          (gas_v2i*)(csr + tb + tid), (las_v2i*)(&s_e[tid]), 0, 0);
#else
      s_e[tid] = csr[tb + tid];
#endif
    }
#if USE_ASYNC_LDS
    wait_async0();
#endif
    __syncthreads();

    const int js = rowStart > tb ? rowStart : tb;
    const int je = rowEnd < tb + cnt ? rowEnd : tb + cnt;
#pragma unroll 2
    for (int j = js; j < je; ++j) {
      const int2 ent = s_e[j - tb];
      const float w = __int_as_float(ent.y);
      const float2 v = *(const float2*)(hcur + (size_t)ent.x * CHANNELS + ch);
      a.x = fmaf(v.x, w, a.x);
      a.y = fmaf(v.y, w, a.y);
    }
  }

  if (valid) {
    const size_t o = (size_t)node * CHANNELS + ch;
    *(float2*)(hnext + o) = a;                 // full overwrite: no pre-zero needed
    float2 av = *(const float2*)(acc + o);
    av.x += a.x; av.y += a.y;
    *(float2*)(acc + o) = av;
  }
}

__global__ void combine_k(const float4* __restrict__ x, const float4* __restrict__ acc,
                          float4* __restrict__ out, int n4) {
  int i = blockIdx.x * blockDim.x + threadIdx.x;
  if (i >= n4) return;
  const float b = (1.0f - ALPHA) / (float)STEPS;  // 0.09
  float4 xv = x[i], av = acc[i], o;
  o.x = ALPHA * xv.x + b * av.x;
  o.y = ALPHA * xv.y + b * av.y;
  o.z = ALPHA * xv.z + b * av.z;
  o.w = ALPHA * xv.w + b * av.w;
  out[i] = o;
}

extern "C" void kernel_launch(void* const* d_in, const int* in_sizes, int n_in,
                              void* d_out, int out_size, void* d_ws, size_t ws_size,
                              hipStream_t stream) {
  const float* x = (const float*)d_in[0];
  const long long* ei = (const long long*)d_in[1];   // int64 edge_index [2,E]
  const int N = in_sizes[0] / CHANNELS;
  const int E = in_sizes[1] / 2;
  float* out = (float*)d_out;
  (void)n_in; (void)out_size; (void)ws_size;

  // ---- workspace carve (256B aligned) ----
  char* ws = (char*)d_ws;
  size_t o = 0;
  auto alignup = [](size_t v) { return (v + 255) & ~(size_t)255; };
  int*   deg    = (int*)(ws + o);    o = alignup(o + (size_t)N * 4);
  int*   off    = (int*)(ws + o);    o = alignup(o + (size_t)(N + 1) * 4);
  int*   cursor = (int*)(ws + o);    o = alignup(o + (size_t)N * 4);
  float* dinv   = (float*)(ws + o);  o = alignup(o + (size_t)N * 4);
  int*   src32  = (int*)(ws + o);    o = alignup(o + (size_t)E * 4);
  int*   dst32  = (int*)(ws + o);    o = alignup(o + (size_t)E * 4);
  int2*  csr    = (int2*)(ws + o);   o = alignup(o + (size_t)E * 8);
  float* h_a    = (float*)(ws + o);  o = alignup(o + (size_t)N * CHANNELS * 4);
  float* h_b    = (float*)(ws + o);  o = alignup(o + (size_t)N * CHANNELS * 4);
  float* acc    = (float*)(ws + o);  o = alignup(o + (size_t)N * CHANNELS * 4);

  const int TB = 256;
  const int nc4 = N * CHANNELS / 4;

  zero_i32 <<<(N + TB - 1) / TB, TB, 0, stream>>>(deg, N);
  zero_f32v4<<<(nc4 + TB - 1) / TB, TB, 0, stream>>>((float4*)acc, nc4);
  deg_convert_k<<<(E + TB - 1) / TB, TB, 0, stream>>>(ei, src32, dst32, deg, E);
  dinv_k   <<<(N + TB - 1) / TB, TB, 0, stream>>>(deg, dinv, N);
  scan_k   <<<1, 1024, 0, stream>>>(deg, off, cursor, N);
  fill_k   <<<(E + TB - 1) / TB, TB, 0, stream>>>(src32, dst32, dinv, cursor, csr, E);

  const int nblocks = (N + WPB - 1) / WPB;
  const float* cur = x;
  float* nxt = h_a;
  for (int t = 0; t < STEPS; ++t) {
    step_k<<<nblocks, TB, 0, stream>>>(cur, nxt, acc, off, csr, N);
    cur = nxt;
    nxt = (nxt == h_a) ? h_b : h_a;
  }
  combine_k<<<(nc4 + TB - 1) / TB, TB, 0, stream>>>((const float4*)x,
                                                    (const float4*)acc,
                                                    (float4*)out, nc4);
}